// ModelNew_80908593922904
// MI455X (gfx1250) — compile-verified
//
#include <hip/hip_runtime.h>
#include <hip/hip_bf16.h>

// Fused conv3d(8->16, k3) + conv_bias + /2 + maxpool2 + spatial-mean + bias + channel-sum.
// Output = 16 floats. Implicit GEMM on V_WMMA_F32_16X16X4_F32 (exact fp32 math):
//   A = weights (16 x 216), B = im2col (216 x 16 w-columns), C = 16ch x 16col.
// One 128-thread block (4 waves) per (batch, d2, h2) pool row; x receptive field
// staged in LDS (32KB) + weights (13.5KB) + k->offset LUT. Everything after the
// conv collapses to a per-wave scalar + one global_atomic_add_f32.

typedef __attribute__((ext_vector_type(2))) float v2f;
typedef __attribute__((ext_vector_type(8))) float v8f;

namespace {
constexpr int DIM    = 64;
constexpr int CIN    = 8;
constexpr int COUT   = 16;
constexpr int KTOT   = CIN * 27;            // 216, divisible by 4
constexpr int SLAB_F = CIN * 4 * 4 * DIM;   // 8192 floats = 32 KB
constexpr int WGT_F  = COUT * KTOT;         // 3456 floats
constexpr float INV_POOL = 1.0f / 29791.0f; // 1 / 31^3
}

__global__ void __launch_bounds__(128)
init_out_kernel(const float* __restrict__ bias, float* __restrict__ out) {
  // out[b] starts at sum_c bias[c]; conv contributions arrive via atomics.
  if (threadIdx.x < 16) {
    float s = 0.0f;
#pragma unroll
    for (int c = 0; c < COUT; ++c) s += bias[c];
    out[threadIdx.x] = s;
  }
}

__global__ void __launch_bounds__(128)
conv_pool_kernel(const float* __restrict__ x, const float* __restrict__ wgt,
                 const float* __restrict__ cbias, float* __restrict__ out) {
  __shared__ float slab[SLAB_F];  // x[b, ci, 2d2+dd(0..3), 2h2+hh(0..3), 0..63]
  __shared__ float lwgt[WGT_F];   // w[cout][k] natural layout
  __shared__ int   loff[KTOT];    // k -> slab element offset (ci*1024+kd*256+kh*64+kw)

  const int d2  = blockIdx.x / 31;
  const int h2  = blockIdx.x % 31;
  const int b   = blockIdx.y;
  const int tid = threadIdx.x;

  // ---- cooperative staging ----
  const size_t xb = (size_t)b * (CIN * DIM * DIM * DIM)
                  + (size_t)(2 * d2) * (DIM * DIM)
                  + (size_t)(2 * h2) * DIM;
  for (int c4 = tid; c4 < SLAB_F / 4; c4 += 128) {
    const int ci = c4 >> 8;
    const int dd = (c4 >> 6) & 3;
    const int hh = (c4 >> 4) & 3;
    const int f4 = c4 & 15;
    const float4 v = *(const float4*)(x + xb + (size_t)ci * (DIM * DIM * DIM)
                                      + (size_t)dd * (DIM * DIM)
                                      + (size_t)hh * DIM + 4 * f4);
    ((float4*)slab)[c4] = v;
  }
  for (int i = tid; i < WGT_F; i += 128) lwgt[i] = wgt[i];
  for (int k = tid; k < KTOT; k += 128) {
    const int ci = k / 27, r = k % 27;
    const int kd = r / 9, r2 = r % 9;
    const int kh = r2 / 3, kw = r2 % 3;
    loff[k] = ci * 1024 + kd * 256 + kh * 64 + kw;
  }
  __syncthreads();

  // ---- per-wave implicit GEMM ----
  const int wave = tid >> 5;       // 4 waves: w-tile q, columns 16q..16q+15
  const int lane = tid & 31;
  const int half = lane >> 4;      // 0: k+{0,1} / ch 0-7 in C ; 1: k+{2,3} / ch 8-15
  const int n    = lane & 15;      // column (B/C) and row M (A)
  const int bcol = (wave << 4) + n;

  v8f c00, c01, c10, c11;          // accumulators per (dr,hr), init with conv_bias
#pragma unroll
  for (int v = 0; v < 8; ++v) {
    const float cv = cbias[8 * half + v];
    c00[v] = cv; c01[v] = cv; c10[v] = cv; c11[v] = cv;
  }

  const int kOff = half << 1;
  for (int t = 0; t < 54; ++t) {
    const int kb = 4 * t + kOff;                       // this lane's k, k+1
    const v2f a = *(const v2f*)&lwgt[n * KTOT + kb];   // A frag: row M=n (ds_load_b64)
    const int2 op = *(const int2*)&loff[kb];           // LUT pair (ds_load_b64)
    const int o0 = op.x + bcol;
    const int o1 = op.y + bcol;
    // B frags for the 4 (dr,hr) combos share addresses up to +dr*256+hr*64.
    v2f b00 = { slab[o0      ], slab[o1      ] };
    v2f b01 = { slab[o0 +  64], slab[o1 +  64] };
    v2f b10 = { slab[o0 + 256], slab[o1 + 256] };
    v2f b11 = { slab[o0 + 320], slab[o1 + 320] };
    c00 = __builtin_amdgcn_wmma_f32_16x16x4_f32(false, a, false, b00, (short)0, c00, false, false);
    c01 = __builtin_amdgcn_wmma_f32_16x16x4_f32(false, a, false, b01, (short)0, c01, false, false);
    c10 = __builtin_amdgcn_wmma_f32_16x16x4_f32(false, a, false, b10, (short)0, c10, false, false);
    c11 = __builtin_amdgcn_wmma_f32_16x16x4_f32(false, a, false, b11, (short)0, c11, false, false);
  }

  // ---- fused maxpool (2x2x2) + mean + channel sum ----
  float s = 0.0f;
#pragma unroll
  for (int v = 0; v < 8; ++v) {
    float m = fmaxf(fmaxf(c00[v], c01[v]), fmaxf(c10[v], c11[v]));  // max over dr,hr
    m = fmaxf(m, __shfl_xor(m, 1, 32));                             // max over w pair
    s += m;                                                         // sum channels in-lane
  }
  // even columns hold window w2 = 8*wave + n/2; mask w2 == 31 (garbage cols 62,63)
  const bool valid = ((n & 1) == 0) && (8 * wave + (n >> 1) < 31);
  s = valid ? s : 0.0f;
  s += __shfl_xor(s, 1, 32);
  s += __shfl_xor(s, 2, 32);
  s += __shfl_xor(s, 4, 32);
  s += __shfl_xor(s, 8, 32);
  s += __shfl_xor(s, 16, 32);
  if (lane == 0) atomicAdd(&out[b], s * (0.5f * INV_POOL));  // /2 and /31^3 folded
}

extern "C" void kernel_launch(void* const* d_in, const int* in_sizes, int n_in,
                              void* d_out, int out_size, void* d_ws, size_t ws_size,
                              hipStream_t stream) {
  const float* x    = (const float*)d_in[0];   // (16,8,64,64,64)
  const float* wgt  = (const float*)d_in[1];   // (16,8,3,3,3)
  const float* cb   = (const float*)d_in[2];   // (16,)
  const float* bias = (const float*)d_in[3];   // (16,1,1,1)
  float* out = (float*)d_out;                  // 16 floats

  init_out_kernel<<<1, 128, 0, stream>>>(bias, out);
  dim3 grid(31 * 31, 16);                      // (d2*31+h2, batch)
  conv_pool_kernel<<<grid, 128, 0, stream>>>(x, wgt, cb, out);
}